// FieldMaker_62234076119695
// MI455X (gfx1250) — compile-verified
//
#include <hip/hip_runtime.h>

typedef __attribute__((ext_vector_type(2))) float v2f;
typedef __attribute__((ext_vector_type(8))) float v8f;

#define NATOMS 512
#define NCH    5
#define ATILES (NATOMS / 16)
#define AST    8   // float stride per preprocessed atom record in LDS (16B-vector aligned)

__device__ __forceinline__ unsigned wave_ballot32(bool p) {
#if defined(__has_builtin)
#if __has_builtin(__builtin_amdgcn_ballot_w32)
  return __builtin_amdgcn_ballot_w32(p);
#else
  return (unsigned)__ballot((int)p);
#endif
#else
  return (unsigned)__ballot((int)p);
#endif
}

// raw v_exp_f32 / v_log_f32 (base-2), no precision-fixup sequences
__device__ __forceinline__ float fast_exp2(float x) {
#if defined(__has_builtin)
#if __has_builtin(__builtin_amdgcn_exp2f)
  return __builtin_amdgcn_exp2f(x);
#else
  return exp2f(x);
#endif
#else
  return exp2f(x);
#endif
}
__device__ __forceinline__ float fast_log2(float x) {
#if defined(__has_builtin)
#if __has_builtin(__builtin_amdgcn_logf)
  return __builtin_amdgcn_logf(x);
#else
  return log2f(x);
#endif
#else
  return log2f(x);
#endif
}

#define LOG2E 1.44269504088896340736f
#define LN2   0.693147180559945309417f

__global__ __launch_bounds__(256) void fieldmaker_wmma_kernel(
    const float* __restrict__ coords,    // [B,512,3]
    const int*   __restrict__ channels,  // [B,512]
    const float* __restrict__ radius,    // [B,512]
    float*       __restrict__ out)       // [B,5,32,32,32]
{
  __shared__ __align__(16) float atomD[NATOMS * AST]; // bx,by,bz,asq, inv,packed
  __shared__ float sC[NATOMS * 3];
  __shared__ float sR[NATOMS];
  __shared__ int   sI[NATOMS];
  __shared__ float acc[8 * 16 * NCH];                 // per-wave 16 voxels x 5 channels

  const int b    = blockIdx.y;
  const int tid  = threadIdx.x;
  const int wave = tid >> 5;
  const int lane = tid & 31;

  // ---- stage raw atom data into LDS via async global->LDS DMA (ASYNCcnt) ----
  const float* gC = coords   + (size_t)b * NATOMS * 3;
  const float* gR = radius   + (size_t)b * NATOMS;
  const int*   gI = channels + (size_t)b * NATOMS;
  for (int i = tid; i < NATOMS * 3; i += 256) {
    unsigned la = (unsigned)(uintptr_t)&sC[i];
    unsigned long long ga = (unsigned long long)(uintptr_t)(gC + i);
    asm volatile("global_load_async_to_lds_b32 %0, %1, off" :: "v"(la), "v"(ga) : "memory");
  }
  for (int i = tid; i < NATOMS; i += 256) {
    unsigned la = (unsigned)(uintptr_t)&sR[i];
    unsigned long long ga = (unsigned long long)(uintptr_t)(gR + i);
    asm volatile("global_load_async_to_lds_b32 %0, %1, off" :: "v"(la), "v"(ga) : "memory");
    unsigned la2 = (unsigned)(uintptr_t)&sI[i];
    unsigned long long ga2 = (unsigned long long)(uintptr_t)(gI + i);
    asm volatile("global_load_async_to_lds_b32 %0, %1, off" :: "v"(la2), "v"(ga2) : "memory");
  }
  asm volatile("s_wait_asynccnt 0x0" ::: "memory");
  __syncthreads();

  // ---- preprocess atoms: scaled coords frame a = 2*coord ----
  for (int a = tid; a < NATOMS; a += 256) {
    float cx = sC[a*3+0], cy = sC[a*3+1], cz = sC[a*3+2];
    int   ch = sI[a];
    float rad = sR[a];
    float ax = 2.f*cx, ay = 2.f*cy, az = 2.f*cz;
    int gx = (int)floorf(ax), gy = (int)floorf(ay), gz = (int)floorf(az);
    float rt  = rad * 2.8284271247461903f;          // radius * sqrt(2) / 0.5
    float inv = 1.0f / (0.8649f * rt * rt);
    int packed = (gx + 64) | ((gy + 64) << 8) | ((gz + 64) << 16) | (ch << 24);
    float* rec = &atomD[a * AST];
    rec[0] = -2.f*ax; rec[1] = -2.f*ay; rec[2] = -2.f*az;
    rec[3] = ax*ax + ay*ay + az*az;
    rec[4] = inv;
    rec[5] = __int_as_float(packed);
  }
  for (int i = tid; i < 8 * 16 * NCH; i += 256) acc[i] = 0.f;
  __syncthreads();

  // ---- this wave's voxel tile: 16 voxels along x, fixed (oy,oz) ----
  const int t     = blockIdx.x * 8 + wave;   // [0, 2048)
  const int xbase = (t & 1) * 16;
  const int oy    = (t >> 1) & 31;
  const int oz    = t >> 6;

  const int half = lane >> 4;
  const int vrow = lane & 15;    // voxel row for A; atom column for B/C/D

  const float px  = (float)(xbase + vrow) - 14.75f;
  const float py  = (float)oy - 14.75f;
  const float pz  = (float)oz - 14.75f;
  const float psq = px*px + py*py + pz*pz;

  // A (16x4 f32): lanes 0-15 -> K={0,1} = (px,py); lanes 16-31 -> K={2,3} = (pz,|p|^2)
  v2f A;
  A.x = half ? pz  : px;
  A.y = half ? psq : py;

  float* wacc = &acc[wave * 16 * NCH];

  for (int tile = 0; tile < ATILES; ++tile) {
    const float4 r0 = *(const float4*)&atomD[(tile*16 + vrow) * AST];
    const float2 r1 = *(const float2*)&atomD[(tile*16 + vrow) * AST + 4];
    const int packed = __float_as_int(r1.y);
    const int gx = (packed & 255) - 64;
    const int gy = ((packed >> 8) & 255) - 64;
    const int gz = ((packed >> 16) & 255) - 64;
    const int ch = (packed >> 24) & 255;

    // cube membership: floor(2*coord) in [o-21, o-11] per axis
    const bool yz = (gy >= oy - 21) && (gy <= oy - 11) &&
                    (gz >= oz - 21) && (gz <= oz - 11);
    const int tx  = gx - xbase - 8 * half;
    const int rlo = tx + 11, rhi = tx + 21;       // valid D-row range
    const bool active = yz && (rhi >= 0) && (rlo <= 7);
    if (wave_ballot32(active) == 0u) continue;    // wave-uniform skip (EXEC stays full)

    // B (4x16 f32): lanes 0-15 -> K={0,1} = (-2ax,-2ay); lanes 16-31 -> (-2az, 1)
    v2f B;
    B.x = half ? r0.z : r0.x;
    B.y = half ? 1.0f : r0.y;
    v8f C;   // |a|^2 per atom column, uniform over rows
    C[0]=r0.w; C[1]=r0.w; C[2]=r0.w; C[3]=r0.w;
    C[4]=r0.w; C[5]=r0.w; C[6]=r0.w; C[7]=r0.w;

    // D[v][a] = |p|^2 - 2 p.a + |a|^2 = dist^2, 256 pairs per instruction
    v8f D = __builtin_amdgcn_wmma_f32_16x16x4_f32(
        false, A, false, B, (short)0, C, false, false);

    const float inv = r1.x;
#pragma unroll
    for (int r = 0; r < 8; ++r) {
      if (active && r >= rlo && r <= rhi) {
        float e = D[r] * inv;
        if (e < 10.0f) {
          // log1p(-exp(-e)) = ln2 * log2(1 - 2^(-e*log2e))
          float val = LN2 * fast_log2(1.0f - fast_exp2(e * -LOG2E));
          atomicAdd(&wacc[(r + 8*half) * NCH + ch], val);   // ds_add_f32
        }
      }
    }
  }

  __syncthreads();

  // ---- finalize: out[b][ch][ox][oy][oz] = 1 - exp(acc) ----
  for (int i = lane; i < 16 * NCH; i += 32) {
    const int m  = i / NCH;
    const int ch = i % NCH;
    const int ox = xbase + m;
    const float v = 1.0f - fast_exp2(wacc[i] * LOG2E);
    out[((((size_t)b * NCH + ch) * 32 + ox) * 32 + oy) * 32 + oz] = v;
  }
}

extern "C" void kernel_launch(void* const* d_in, const int* in_sizes, int n_in,
                              void* d_out, int out_size, void* d_ws, size_t ws_size,
                              hipStream_t stream) {
  (void)n_in; (void)out_size; (void)d_ws; (void)ws_size;
  const float* coords   = (const float*)d_in[0];
  const int*   channels = (const int*)d_in[1];
  const float* radius   = (const float*)d_in[2];
  float* out = (float*)d_out;

  const int B = in_sizes[1] / NATOMS;   // 32
  dim3 grid(256, B);                    // 256 blocks x 8 waves = 2048 voxel tiles/batch
  fieldmaker_wmma_kernel<<<grid, 256, 0, stream>>>(coords, channels, radius, out);
}